// GCN_36687610642609
// MI455X (gfx1250) — compile-verified
//
#include <hip/hip_runtime.h>
#include <math.h>

#define GN 100000
#define GE 1600000
#define NFEAT 256
#define NHID 128
#define NCLASS 47
#define NCPAD 48
#define LW1_STRIDE 136   // padded row stride (floats) for W1 LDS tile

typedef __attribute__((ext_vector_type(2))) float v2f;
typedef __attribute__((ext_vector_type(8))) float v8f;

// ---------------------------------------------------------------------------
// Zero-fill
// ---------------------------------------------------------------------------
__global__ void zero_f32(float* __restrict__ p, size_t n) {
    size_t i = (size_t)blockIdx.x * blockDim.x + threadIdx.x;
    if (i < n) p[i] = 0.0f;
}

// ---------------------------------------------------------------------------
// GEMM1: h0[N,128] = x[N,256] @ W1[256,128]   (fp32 WMMA 16x16x4)
// Block = 256 thr = 8 waves. W1 staged in LDS once per block (dynamic, 139KB).
// Each wave computes a full 16-row x 128-col stripe: 8 accumulators,
// 8 WMMAs per k-step, A-fragment loaded once per k-step.
// ---------------------------------------------------------------------------
__global__ void gemm1_wmma(const float* __restrict__ x,
                           const float* __restrict__ W1,
                           float* __restrict__ h0) {
    extern __shared__ float lw[];             // [256][LW1_STRIDE]

    const int tid = threadIdx.x;
    // Cooperative W1 load: 8192 float4, coalesced; 16B-aligned LDS stores.
    #pragma unroll 4
    for (int i4 = tid; i4 < (NFEAT * NHID) / 4; i4 += 256) {
        const int k = i4 >> 5;                // 32 float4 per 128-wide row
        const int col = (i4 & 31) * 4;
        *(float4*)&lw[k * LW1_STRIDE + col] = ((const float4*)W1)[i4];
    }
    __syncthreads();

    const int wave = __builtin_amdgcn_readfirstlane(tid >> 5);
    const int lane = tid & 31;
    const int tile = blockIdx.x * 8 + wave;   // 16-row tile id
    if (tile >= GN / 16) return;

    const int row0 = tile * 16;
    const int half = lane >> 4;               // 0: lanes 0-15, 1: lanes 16-31
    const int l16  = lane & 15;

    // A fragment: lanes 0-15 = rows, K=k,k+1; lanes 16-31: K=k+2,k+3
    const float* arow = x + (size_t)(row0 + l16) * NFEAT + 2 * half;
    // B fragment base in LDS: row (K) = k + 2*half, col = t*16 + l16
    const float* bbase = lw + (size_t)(2 * half) * LW1_STRIDE + l16;

    v8f acc[8];
    #pragma unroll
    for (int t = 0; t < 8; ++t) acc[t] = (v8f){};

    for (int k = 0; k < NFEAT; k += 4) {
        v2f a;
        a.x = arow[k];
        a.y = arow[k + 1];
        const float* bk0 = bbase + (size_t)k * LW1_STRIDE;
        const float* bk1 = bk0 + LW1_STRIDE;
        #pragma unroll
        for (int t = 0; t < 8; ++t) {
            v2f b;
            b.x = bk0[t * 16];
            b.y = bk1[t * 16];
            acc[t] = __builtin_amdgcn_wmma_f32_16x16x4_f32(
                         false, a, false, b, (short)0, acc[t], false, false);
        }
    }

    // C/D layout: VGPR r -> rows {row0+r (lanes 0-15), row0+8+r (lanes 16-31)}
    float* drow = h0 + (size_t)(row0 + half * 8) * NHID + l16;
    #pragma unroll
    for (int t = 0; t < 8; ++t)
        #pragma unroll
        for (int r = 0; r < 8; ++r)
            drow[(size_t)r * NHID + t * 16] = acc[t][r];
}

// ---------------------------------------------------------------------------
// SpMM1: agg[row[e]] += val[e] * h0[col[e]]   (128 feats, one wave per edge)
// ---------------------------------------------------------------------------
__global__ void spmm1_atomic(const int* __restrict__ erow,
                             const int* __restrict__ ecol,
                             const float* __restrict__ eval,
                             const float* __restrict__ h0,
                             float* __restrict__ agg) {
    const int w    = __builtin_amdgcn_readfirstlane(threadIdx.x >> 5);
    const int e    = blockIdx.x * 8 + w;      // wave-uniform (scalar) edge id
    const int lane = threadIdx.x & 31;
    if (e >= GE) return;
    const int   r    = erow[e];
    const int   cidx = ecol[e];
    const float v    = eval[e];

    const float4 d = ((const float4*)(h0 + (size_t)cidx * NHID))[lane];
    float* dst = agg + (size_t)r * NHID + lane * 4;
    atomicAdd(dst + 0, v * d.x);
    atomicAdd(dst + 1, v * d.y);
    atomicAdd(dst + 2, v * d.z);
    atomicAdd(dst + 3, v * d.w);
}

// ---------------------------------------------------------------------------
// h = relu(agg + b1)
// ---------------------------------------------------------------------------
__global__ void bias_relu(float* __restrict__ agg, const float* __restrict__ b1,
                          size_t total) {
    size_t i = (size_t)blockIdx.x * blockDim.x + threadIdx.x;
    if (i < total) {
        float t = agg[i] + b1[i & (NHID - 1)];
        agg[i] = t > 0.0f ? t : 0.0f;
    }
}

// ---------------------------------------------------------------------------
// GEMM2: o0[N,48] = h[N,128] @ W2[128,47]  (cols padded to 48)
// Block = 256 thr = 8 waves; W2 zero-padded into static LDS (24KB).
// Each wave computes a 16-row x 48-col stripe (3 accumulators).
// ---------------------------------------------------------------------------
__global__ void gemm2_wmma(const float* __restrict__ h,
                           const float* __restrict__ W2,
                           float* __restrict__ o0) {
    __shared__ float lw2[NHID * NCPAD];       // [128][48]

    const int tid = threadIdx.x;
    #pragma unroll 4
    for (int i = tid; i < NHID * NCPAD; i += 256) {
        const int k = i / NCPAD;
        const int c = i - k * NCPAD;
        lw2[i] = (c < NCLASS) ? W2[(size_t)k * NCLASS + c] : 0.0f;
    }
    __syncthreads();

    const int wave = __builtin_amdgcn_readfirstlane(tid >> 5);
    const int lane = tid & 31;
    const int tile = blockIdx.x * 8 + wave;
    if (tile >= GN / 16) return;

    const int row0 = tile * 16;
    const int half = lane >> 4;
    const int l16  = lane & 15;

    const float* arow  = h + (size_t)(row0 + l16) * NHID + 2 * half;
    const float* bbase = lw2 + (size_t)(2 * half) * NCPAD + l16;

    v8f acc[3];
    #pragma unroll
    for (int t = 0; t < 3; ++t) acc[t] = (v8f){};

    for (int k = 0; k < NHID; k += 4) {
        v2f a;
        a.x = arow[k];
        a.y = arow[k + 1];
        const float* bk0 = bbase + (size_t)k * NCPAD;
        const float* bk1 = bk0 + NCPAD;
        #pragma unroll
        for (int t = 0; t < 3; ++t) {
            v2f b;
            b.x = bk0[t * 16];
            b.y = bk1[t * 16];
            acc[t] = __builtin_amdgcn_wmma_f32_16x16x4_f32(
                         false, a, false, b, (short)0, acc[t], false, false);
        }
    }

    float* drow = o0 + (size_t)(row0 + half * 8) * NCPAD + l16;
    #pragma unroll
    for (int t = 0; t < 3; ++t)
        #pragma unroll
        for (int r = 0; r < 8; ++r)
            drow[(size_t)r * NCPAD + t * 16] = acc[t][r];
}

// ---------------------------------------------------------------------------
// SpMM2: out[row[e]] += val[e] * o0[col[e]]   (47 feats, one wave per edge)
// ---------------------------------------------------------------------------
__global__ void spmm2_atomic(const int* __restrict__ erow,
                             const int* __restrict__ ecol,
                             const float* __restrict__ eval,
                             const float* __restrict__ o0,
                             float* __restrict__ out) {
    const int w    = __builtin_amdgcn_readfirstlane(threadIdx.x >> 5);
    const int e    = blockIdx.x * 8 + w;
    const int lane = threadIdx.x & 31;
    if (e >= GE) return;
    const int   r    = erow[e];
    const int   cidx = ecol[e];
    const float v    = eval[e];

    const float* src = o0 + (size_t)cidx * NCPAD;
    float*       dst = out + (size_t)r * NCLASS;
    atomicAdd(dst + lane, v * src[lane]);            // lanes 0..31 < 47
    const int f2 = lane + 32;
    if (f2 < NCLASS) atomicAdd(dst + f2, v * src[f2]);
}

// ---------------------------------------------------------------------------
// out = log_softmax(out + b2), one wave per row, wave32 shuffle reductions
// ---------------------------------------------------------------------------
__global__ void bias_logsoftmax(float* __restrict__ out,
                                const float* __restrict__ b2) {
    const int w    = __builtin_amdgcn_readfirstlane(threadIdx.x >> 5);
    const int row  = blockIdx.x * 8 + w;
    const int lane = threadIdx.x & 31;
    if (row >= GN) return;
    float* p = out + (size_t)row * NCLASS;

    const int  f2 = lane + 32;
    const bool k2 = (f2 < NCLASS);
    float v0 = p[lane] + b2[lane];
    float v1 = k2 ? (p[f2] + b2[f2]) : -INFINITY;

    float m = fmaxf(v0, v1);
    #pragma unroll
    for (int off = 16; off > 0; off >>= 1)
        m = fmaxf(m, __shfl_xor(m, off, 32));

    float s = __expf(v0 - m) + (k2 ? __expf(v1 - m) : 0.0f);
    #pragma unroll
    for (int off = 16; off > 0; off >>= 1)
        s += __shfl_xor(s, off, 32);

    const float ls = __logf(s);
    p[lane] = v0 - m - ls;
    if (k2) p[f2] = v1 - m - ls;
}

// ---------------------------------------------------------------------------
extern "C" void kernel_launch(void* const* d_in, const int* in_sizes, int n_in,
                              void* d_out, int out_size, void* d_ws, size_t ws_size,
                              hipStream_t stream) {
    const float* x    = (const float*)d_in[0];
    const int*   erow = (const int*)  d_in[1];
    const int*   ecol = (const int*)  d_in[2];
    const float* eval = (const float*)d_in[3];
    const float* W1   = (const float*)d_in[4];
    const float* b1   = (const float*)d_in[5];
    const float* W2   = (const float*)d_in[6];
    const float* b2   = (const float*)d_in[7];
    float* out = (float*)d_out;
    float* ws  = (float*)d_ws;

    float* h0   = ws;                        // [N,128]  (reused as o0 later)
    float* agg1 = ws + (size_t)GN * NHID;    // [N,128]
    float* o0   = ws;                        // [N,48], recycled h0 region

    const size_t n_h   = (size_t)GN * NHID;  // 12.8M
    const size_t n_out = (size_t)GN * NCLASS;

    const int ntiles      = GN / 16;                    // 6250
    const int gemm_blocks = (ntiles + 7) / 8;           // 782
    const size_t lds1     = (size_t)NFEAT * LW1_STRIDE * sizeof(float);

    zero_f32<<<(int)((n_h   + 255) / 256), 256, 0, stream>>>(agg1, n_h);
    zero_f32<<<(int)((n_out + 255) / 256), 256, 0, stream>>>(out, n_out);

    gemm1_wmma<<<gemm_blocks, 256, lds1, stream>>>(x, W1, h0);
    spmm1_atomic<<<GE / 8, 256, 0, stream>>>(erow, ecol, eval, h0, agg1);
    bias_relu<<<(int)((n_h + 255) / 256), 256, 0, stream>>>(agg1, b1, n_h);

    gemm2_wmma<<<gemm_blocks, 256, 0, stream>>>(agg1, W2, o0);
    spmm2_atomic<<<GE / 8, 256, 0, stream>>>(erow, ecol, eval, o0, out);
    bias_logsoftmax<<<(GN * 32 + 255) / 256, 256, 0, stream>>>(out, b2);
}